// GAT_85023172592276
// MI455X (gfx1250) — compile-verified
//
#include <hip/hip_runtime.h>
#include <math.h>

typedef float v2f __attribute__((ext_vector_type(2)));
typedef float v8f __attribute__((ext_vector_type(8)));

#define GAT_HEADS 4
#define GAT_HID   64
#define GAT_CH    256   // HEADS*HID

// ---------------------------------------------------------------------------
// Fold attention vector into projection weights:
// out[k*4+h] = sum_c w[k*256 + h*64 + c] * att[h*64 + c]
// ---------------------------------------------------------------------------
__global__ void fold_att_kernel(const float* __restrict__ w,
                                const float* __restrict__ att,
                                float* __restrict__ out, int K) {
  int i = blockIdx.x * blockDim.x + threadIdx.x;
  if (i >= K * GAT_HEADS) return;
  int k = i >> 2, h = i & 3;
  float s = 0.f;
  #pragma unroll 16
  for (int c = 0; c < GAT_HID; ++c)
    s += w[(size_t)k * GAT_CH + h * GAT_HID + c] * att[h * GAT_HID + c];
  out[k * GAT_HEADS + h] = s;
}

// ---------------------------------------------------------------------------
// WMMA fp32 GEMM: C[M,N] = A[M,K] @ B[K,N], row-major, K % 4 == 0.
// One wave computes a 16 x (16*NT) output strip via V_WMMA_F32_16X16X4_F32:
// the A fragment is loaded once per K-step (single b64 load) and fed to NT
// back-to-back WMMAs. Out-of-range rows/cols are handled by INDEX CLAMPING
// (a garbage A-row m only pollutes D-row m; garbage B-col n only pollutes
// D-col n — both skipped by the store guards), so all loads are branchless.
//
// Fragment layouts (ISA 7.12.2):
//   A 16x4 : lanes 0-15 -> M=lane, VGPR0/1 = K0/K1; lanes 16-31 -> K2/K3
//   B 4x16 : mirrored (lane -> N, half-wave selects K pair)
//   C 16x16: VGPR r -> M=r (lanes 0-15) / M=r+8 (lanes 16-31), N = lane&15
// ---------------------------------------------------------------------------
template <int NT>
__global__ void gemm_wmma_f32(const float* __restrict__ A,
                              const float* __restrict__ B,
                              float* __restrict__ C,
                              int M, int K, int N) {
  const int lane = threadIdx.x & 31;
  const int Mtiles = (M + 15) >> 4;
  const int NG = (N + 16 * NT - 1) / (16 * NT);
  const long long tile =
      (long long)blockIdx.x * (blockDim.x >> 5) + (threadIdx.x >> 5);
  if (tile >= (long long)Mtiles * NG) return;  // wave-uniform exit
  const int mt = (int)(tile % Mtiles);
  const int ng = (int)(tile / Mtiles);

  const int kh  = lane >> 4;   // which K-pair this half-wave holds
  const int l15 = lane & 15;

  int am = mt * 16 + l15;
  if (am >= M) am = M - 1;     // clamp (store-guarded rows only)

  int  bn[NT];
  bool bok[NT];
  #pragma unroll
  for (int t = 0; t < NT; ++t) {
    int n  = ng * 16 * NT + t * 16 + l15;
    bok[t] = (n < N);
    bn[t]  = bok[t] ? n : (N - 1);  // clamp (store-guarded cols only)
  }

  const float* ap = A + (size_t)am * K + kh * 2;       // 8B aligned (K even)
  const float* bp = B + (size_t)(kh * 2) * N;

  v8f acc[NT];
  #pragma unroll
  for (int t = 0; t < NT; ++t) acc[t] = (v8f){};

  for (int k0 = 0; k0 < K; k0 += 4) {
    v2f a = *(const v2f*)ap;   // global_load_b64
    #pragma unroll
    for (int t = 0; t < NT; ++t) {
      v2f b;
      b.x = bp[bn[t]];
      b.y = bp[bn[t] + N];
      acc[t] = __builtin_amdgcn_wmma_f32_16x16x4_f32(
          false, a, false, b, (short)0, acc[t], false, false);
    }
    ap += 4;
    bp += (size_t)4 * N;
  }

  #pragma unroll
  for (int t = 0; t < NT; ++t) {
    if (!bok[t]) continue;
    #pragma unroll
    for (int r = 0; r < 8; ++r) {
      int cm = mt * 16 + r + kh * 8;
      if (cm < M) C[(size_t)cm * N + bn[t]] = acc[t][r];
    }
  }
}

// ---------------------------------------------------------------------------
// Fills
// ---------------------------------------------------------------------------
__global__ void fill_u32(unsigned* p, unsigned v, long long n) {
  long long i = (long long)blockIdx.x * blockDim.x + threadIdx.x;
  if (i < n) p[i] = v;
}
__global__ void fill_f32(float* p, float v, long long n) {
  long long i = (long long)blockIdx.x * blockDim.x + threadIdx.x;
  if (i < n) p[i] = v;
}

// ordered-uint encoding of float for atomic max (total order incl. negatives)
__device__ __forceinline__ unsigned flip_f32(float f) {
  unsigned u = __float_as_uint(f);
  return (u & 0x80000000u) ? ~u : (u | 0x80000000u);
}
__device__ __forceinline__ float unflip_f32(unsigned u) {
  u = (u & 0x80000000u) ? (u & 0x7FFFFFFFu) : ~u;
  return __uint_as_float(u);
}

// ---------------------------------------------------------------------------
// Edge pass A: alpha = leaky_relu(al_s[src] + al_d[dst]); atomic max per dst
// ---------------------------------------------------------------------------
__global__ void edge_alpha_kernel(const int* __restrict__ src,
                                  const int* __restrict__ dst,
                                  const float* __restrict__ als,
                                  const float* __restrict__ ald,
                                  float* __restrict__ alphaE,
                                  unsigned* __restrict__ amax,
                                  long long EH) {
  long long i = (long long)blockIdx.x * blockDim.x + threadIdx.x;
  if (i >= EH) return;
  int e = (int)(i >> 2), h = (int)(i & 3);
  int s = src[e], d = dst[e];
  float a = als[(size_t)s * 4 + h] + ald[(size_t)d * 4 + h];
  a = (a > 0.f) ? a : 0.2f * a;           // leaky_relu, slope 0.2
  alphaE[i] = a;
  atomicMax(&amax[(size_t)d * 4 + h], flip_f32(a));
}

// ---------------------------------------------------------------------------
// Edge pass B: w = exp(alpha - amax[dst]); atomic sum into denom[dst]
// ---------------------------------------------------------------------------
__global__ void edge_exp_kernel(const int* __restrict__ dst,
                                float* __restrict__ alphaE,
                                const unsigned* __restrict__ amax,
                                float* __restrict__ denom,
                                long long EH) {
  long long i = (long long)blockIdx.x * blockDim.x + threadIdx.x;
  if (i >= EH) return;
  int e = (int)(i >> 2), h = (int)(i & 3);
  int d = dst[e];
  float m = unflip_f32(amax[(size_t)d * 4 + h]);
  float w = expf(alphaE[i] - m);
  alphaE[i] = w;
  atomicAdd(&denom[(size_t)d * 4 + h], w);
}

// ---------------------------------------------------------------------------
// Edge pass C: agg[dst, c] += hs[src, c] * alpha / denom[dst]
// One 256-thread block per edge -> coalesced channel reads, L2 atomics.
// ---------------------------------------------------------------------------
__global__ void edge_aggregate_kernel(const int* __restrict__ src,
                                      const int* __restrict__ dst,
                                      const float* __restrict__ alphaE,
                                      const float* __restrict__ denom,
                                      const float* __restrict__ hs,
                                      float* __restrict__ agg,
                                      long long EC) {
  long long i = (long long)blockIdx.x * blockDim.x + threadIdx.x;
  if (i >= EC) return;
  int e = (int)(i >> 8);
  int c = (int)(i & 255);
  int h = c >> 6;
  int s = src[e], d = dst[e];
  float coeff = alphaE[(size_t)e * 4 + h] / denom[(size_t)d * 4 + h];
  atomicAdd(&agg[(size_t)d * GAT_CH + c],
            hs[(size_t)s * GAT_CH + c] * coeff);
}

// ---------------------------------------------------------------------------
// Layer-0 finalize: h = ELU(agg + b0 + bsk0), in place
// ---------------------------------------------------------------------------
__global__ void finalize0_kernel(float* __restrict__ agg,
                                 const float* __restrict__ b0,
                                 const float* __restrict__ bsk0,
                                 long long total) {
  long long i = (long long)blockIdx.x * blockDim.x + threadIdx.x;
  if (i >= total) return;
  int c = (int)(i & 255);
  float v = agg[i] + b0[c] + bsk0[c];
  agg[i] = (v > 0.f) ? v : expm1f(v);
}

// ---------------------------------------------------------------------------
// Layer-1 finalize + log_softmax.
// o[n,c] = mean_h agg1[n,h,c] + b1[c] + skip1[n,c] + bsk1[c]; log_softmax(c).
// One wave (32 lanes) per node, 2 channels per lane, shuffle reductions.
// ---------------------------------------------------------------------------
__global__ void finalize1_logsoftmax_kernel(const float* __restrict__ agg1,
                                            const float* __restrict__ skip1,
                                            const float* __restrict__ b1,
                                            const float* __restrict__ bsk1,
                                            float* __restrict__ out, int Nn) {
  int n = blockIdx.x;
  if (n >= Nn) return;
  int t = threadIdx.x;  // 0..31, wave32
  float v[2];
  #pragma unroll
  for (int j = 0; j < 2; ++j) {
    int c = t + j * 32;
    float s = 0.f;
    #pragma unroll
    for (int h = 0; h < GAT_HEADS; ++h)
      s += agg1[(size_t)n * GAT_CH + h * GAT_HID + c];
    v[j] = 0.25f * s + b1[c] + skip1[(size_t)n * GAT_HID + c] + bsk1[c];
  }
  float m = fmaxf(v[0], v[1]);
  #pragma unroll
  for (int off = 16; off > 0; off >>= 1)
    m = fmaxf(m, __shfl_xor(m, off, 32));
  float s = expf(v[0] - m) + expf(v[1] - m);
  #pragma unroll
  for (int off = 16; off > 0; off >>= 1)
    s += __shfl_xor(s, off, 32);
  float ls = logf(s);
  out[(size_t)n * GAT_HID + t]      = v[0] - m - ls;
  out[(size_t)n * GAT_HID + t + 32] = v[1] - m - ls;
}

// ---------------------------------------------------------------------------
// Host-side launcher
// ---------------------------------------------------------------------------
static inline long long cdiv_ll(long long a, long long b) { return (a + b - 1) / b; }

extern "C" void kernel_launch(void* const* d_in, const int* in_sizes, int n_in,
                              void* d_out, int out_size, void* d_ws, size_t ws_size,
                              hipStream_t stream) {
  const float* x        = (const float*)d_in[0];
  const int*   ei0      = (const int*)d_in[1];
  const int*   ei1      = (const int*)d_in[2];
  const float* w_src0   = (const float*)d_in[3];
  const float* w_dst0   = (const float*)d_in[4];
  const float* att_src0 = (const float*)d_in[5];
  const float* att_dst0 = (const float*)d_in[6];
  const float* b0       = (const float*)d_in[7];
  const float* wsk0     = (const float*)d_in[8];
  const float* bsk0     = (const float*)d_in[9];
  const float* w_src1   = (const float*)d_in[10];
  const float* w_dst1   = (const float*)d_in[11];
  const float* att_src1 = (const float*)d_in[12];
  const float* att_dst1 = (const float*)d_in[13];
  const float* b1       = (const float*)d_in[14];
  const float* wsk1     = (const float*)d_in[15];
  const float* bsk1     = (const float*)d_in[16];
  float* out = (float*)d_out;

  const int N0 = 500000, N1 = 50000, N2 = 5000;
  const int K0 = 128, CH = GAT_CH, OUT = GAT_HID;
  const int E0 = in_sizes[1] / 2;
  const int E1 = in_sizes[2] / 2;
  const int* src0 = ei0;       const int* dst0 = ei0 + E0;
  const int* src1 = ei1;       const int* dst1 = ei1 + E1;

  // ---- workspace bump allocator (256B aligned) ----
  char*  ws  = (char*)d_ws;
  size_t off = 0;
  auto alloc = [&](size_t bytes) -> void* {
    void* p = ws + off;
    off += (bytes + 255) & ~(size_t)255;
    return p;
  };
  float*    hs0     = (float*)alloc((size_t)N0 * CH * 4);   // 512 MB
  float*    agg0    = (float*)alloc((size_t)N1 * CH * 4);   // 51 MB (becomes h)
  float*    hs1     = (float*)alloc((size_t)N1 * CH * 4);   // 51 MB
  float*    agg1    = (float*)alloc((size_t)N2 * CH * 4);
  float*    als0    = (float*)alloc((size_t)N0 * 4 * 4);
  float*    ald0    = (float*)alloc((size_t)N1 * 4 * 4);
  unsigned* amax0   = (unsigned*)alloc((size_t)N1 * 4 * 4);
  float*    denom0  = (float*)alloc((size_t)N1 * 4 * 4);
  float*    alphaE0 = (float*)alloc((size_t)E0 * 4 * 4);
  float*    als1    = (float*)alloc((size_t)N1 * 4 * 4);
  float*    ald1    = (float*)alloc((size_t)N2 * 4 * 4);
  unsigned* amax1   = (unsigned*)alloc((size_t)N2 * 4 * 4);
  float*    denom1  = (float*)alloc((size_t)N2 * 4 * 4);
  float*    alphaE1 = (float*)alloc((size_t)E1 * 4 * 4);
  float*    skip1   = (float*)alloc((size_t)N2 * OUT * 4);
  float*    wf_s0   = (float*)alloc((size_t)K0 * 4 * 4);
  float*    wf_d0   = (float*)alloc((size_t)K0 * 4 * 4);
  float*    wf_s1   = (float*)alloc((size_t)CH * 4 * 4);
  float*    wf_d1   = (float*)alloc((size_t)CH * 4 * 4);
  (void)ws_size; (void)n_in; (void)out_size; (void)w_dst0; (void)w_dst1;

  const int TB = 256;                 // 8 waves per block
  const unsigned NEG_INF_FLIP = 0x007FFFFFu;  // flip(-inf)

  // ---- fold attention vectors into projection weights ----
  fold_att_kernel<<<cdiv_ll((long long)K0 * 4, TB), TB, 0, stream>>>(w_src0, att_src0, wf_s0, K0);
  fold_att_kernel<<<cdiv_ll((long long)K0 * 4, TB), TB, 0, stream>>>(w_dst0, att_dst0, wf_d0, K0);
  fold_att_kernel<<<cdiv_ll((long long)CH * 4, TB), TB, 0, stream>>>(w_src1, att_src1, wf_s1, CH);
  fold_att_kernel<<<cdiv_ll((long long)CH * 4, TB), TB, 0, stream>>>(w_dst1, att_dst1, wf_d1, CH);

  // wide GEMMs (N multiple of 64): 16x64 strip per wave, 4 WMMAs per K-step
  auto gemm4 = [&](const float* A, const float* B, float* C, int M, int K, int N) {
    long long tiles = (long long)((M + 15) / 16) * ((N + 63) / 64);
    gemm_wmma_f32<4><<<(unsigned)cdiv_ll(tiles, TB / 32), TB, 0, stream>>>(A, B, C, M, K, N);
  };
  // narrow GEMMs (N = 4): single 16x16 tile per wave
  auto gemm1 = [&](const float* A, const float* B, float* C, int M, int K, int N) {
    long long tiles = (long long)((M + 15) / 16) * ((N + 15) / 16);
    gemm_wmma_f32<1><<<(unsigned)cdiv_ll(tiles, TB / 32), TB, 0, stream>>>(A, B, C, M, K, N);
  };

  // ================= Layer 0 =================
  gemm4(x, w_src0, hs0, N0, K0, CH);        // hs0 = x @ w_src0   (big WMMA GEMM)
  gemm1(x, wf_s0,  als0, N0, K0, 4);        // al_s (folded)
  gemm1(x, wf_d0,  ald0, N1, K0, 4);        // al_d (folded, dst nodes only)
  gemm4(x, wsk0,   agg0, N1, K0, CH);       // agg0 initialized with skip: xt @ wsk0

  fill_u32<<<cdiv_ll((long long)N1 * 4, TB), TB, 0, stream>>>(amax0, NEG_INF_FLIP, (long long)N1 * 4);
  fill_f32<<<cdiv_ll((long long)N1 * 4, TB), TB, 0, stream>>>(denom0, 0.f, (long long)N1 * 4);

  long long EH0 = (long long)E0 * 4;
  edge_alpha_kernel<<<cdiv_ll(EH0, TB), TB, 0, stream>>>(src0, dst0, als0, ald0, alphaE0, amax0, EH0);
  edge_exp_kernel<<<cdiv_ll(EH0, TB), TB, 0, stream>>>(dst0, alphaE0, amax0, denom0, EH0);
  long long EC0 = (long long)E0 * CH;
  edge_aggregate_kernel<<<cdiv_ll(EC0, TB), TB, 0, stream>>>(src0, dst0, alphaE0, denom0, hs0, agg0, EC0);

  finalize0_kernel<<<cdiv_ll((long long)N1 * CH, TB), TB, 0, stream>>>(agg0, b0, bsk0, (long long)N1 * CH);

  // ================= Layer 1 =================
  gemm4(agg0, w_src1, hs1, N1, CH, CH);     // hs1 = h @ w_src1
  gemm1(agg0, wf_s1,  als1, N1, CH, 4);
  gemm1(agg0, wf_d1,  ald1, N2, CH, 4);
  gemm4(agg0, wsk1,   skip1, N2, CH, OUT);  // skip1 = ht @ wsk1

  fill_u32<<<cdiv_ll((long long)N2 * 4, TB), TB, 0, stream>>>(amax1, NEG_INF_FLIP, (long long)N2 * 4);
  fill_f32<<<cdiv_ll((long long)N2 * 4, TB), TB, 0, stream>>>(denom1, 0.f, (long long)N2 * 4);
  fill_f32<<<cdiv_ll((long long)N2 * CH, TB), TB, 0, stream>>>(agg1, 0.f, (long long)N2 * CH);

  long long EH1 = (long long)E1 * 4;
  edge_alpha_kernel<<<cdiv_ll(EH1, TB), TB, 0, stream>>>(src1, dst1, als1, ald1, alphaE1, amax1, EH1);
  edge_exp_kernel<<<cdiv_ll(EH1, TB), TB, 0, stream>>>(dst1, alphaE1, amax1, denom1, EH1);
  long long EC1 = (long long)E1 * CH;
  edge_aggregate_kernel<<<cdiv_ll(EC1, TB), TB, 0, stream>>>(src1, dst1, alphaE1, denom1, hs1, agg1, EC1);

  finalize1_logsoftmax_kernel<<<N2, 32, 0, stream>>>(agg1, skip1, b1, bsk1, out, N2);
}